// GCNGraphClassifier_84490596647536
// MI455X (gfx1250) — compile-verified
//
#include <hip/hip_runtime.h>
#include <hip/hip_bf16.h>

typedef __attribute__((ext_vector_type(2))) float v2f;
typedef __attribute__((ext_vector_type(8))) float v8f;

#define HIDC 64
#define NCLS 10
#define NGRAPH 64
#define WPAD 68   // padded LDS stride for transposed W (conflict-free b64 reads, 8B aligned)

// ---------------------------------------------------------------------------
// fill with zeros
// ---------------------------------------------------------------------------
__global__ void fill0_kernel(float* __restrict__ p, int n) {
    int i = blockIdx.x * blockDim.x + threadIdx.x;
    if (i < n) p[i] = 0.0f;
}

// ---------------------------------------------------------------------------
// degree accumulation on dst (self-loop +1 added in dinv kernel)
// ---------------------------------------------------------------------------
__global__ void deg_kernel(const int* __restrict__ dst, float* __restrict__ deg, int E) {
    int e = blockIdx.x * blockDim.x + threadIdx.x;
    if (e < E) atomicAdd(&deg[dst[e]], 1.0f);
}

// deg -> dinv (in place) and self_norm = dinv^2
__global__ void dinv_kernel(float* __restrict__ deg_dinv, float* __restrict__ self_norm, int N) {
    int i = blockIdx.x * blockDim.x + threadIdx.x;
    if (i < N) {
        float dv = 1.0f / sqrtf(deg_dinv[i] + 1.0f);   // deg >= 1 always
        deg_dinv[i]  = dv;
        self_norm[i] = dv * dv;
    }
}

__global__ void edge_norm_kernel(const int* __restrict__ src, const int* __restrict__ dst,
                                 const float* __restrict__ dinv, float* __restrict__ en, int E) {
    int e = blockIdx.x * blockDim.x + threadIdx.x;
    if (e < E) en[e] = dinv[src[e]] * dinv[dst[e]];
}

// ---------------------------------------------------------------------------
// H = X @ W  (N x 64) @ (64 x 64), f32 WMMA 16x16x4.
// 256 threads = 8 waves per block; wave w computes rows [ (blk*8+w)*16, +16 ).
// W is staged TRANSPOSED in LDS with stride WPAD so each B fragment
// (elements (k,j),(k+1,j)) is one aligned ds_load_b64 straight into an
// even-aligned VGPR pair (WMMA source requirement) -- no repacking movs.
// ---------------------------------------------------------------------------
__global__ __launch_bounds__(256) void gemm64_wmma_kernel(
        const float* __restrict__ X, const float* __restrict__ W,
        float* __restrict__ H, int N) {
    __shared__ float Wt[HIDC * WPAD];
    for (int i = threadIdx.x; i < HIDC * HIDC; i += 256) {
        int k = i >> 6;          // input channel (K)
        int j = i & 63;          // output channel (N)
        Wt[j * WPAD + k] = W[i];
    }
    __syncthreads();

    const int wave = threadIdx.x >> 5;
    const int lane = threadIdx.x & 31;
    const int m    = lane & 15;            // M index (A) / N index (B,C)
    const int kh   = (lane >> 4) << 1;     // K sub-offset: 0 for lanes 0-15, 2 for 16-31

    const int rowBase = (blockIdx.x * 8 + wave) * 16;
    if (rowBase >= N) return;              // wave-uniform: EXEC all-1s inside WMMA

    const float* __restrict__ Xr = X + (size_t)(rowBase + m) * HIDC + kh;
    const float* __restrict__ Wb = Wt + m * WPAD + kh;   // + jt*16*WPAD + kt*4

    v8f acc0 = {}, acc1 = {}, acc2 = {}, acc3 = {};
#pragma unroll
    for (int kt = 0; kt < 16; ++kt) {
        const int k4 = kt * 4;
        const v2f a  = *(const v2f*)(Xr + k4);
        const v2f b0 = *(const v2f*)(Wb + k4);
        const v2f b1 = *(const v2f*)(Wb + 16 * WPAD + k4);
        const v2f b2 = *(const v2f*)(Wb + 32 * WPAD + k4);
        const v2f b3 = *(const v2f*)(Wb + 48 * WPAD + k4);
        acc0 = __builtin_amdgcn_wmma_f32_16x16x4_f32(false, a, false, b0, (short)0, acc0, false, false);
        acc1 = __builtin_amdgcn_wmma_f32_16x16x4_f32(false, a, false, b1, (short)0, acc1, false, false);
        acc2 = __builtin_amdgcn_wmma_f32_16x16x4_f32(false, a, false, b2, (short)0, acc2, false, false);
        acc3 = __builtin_amdgcn_wmma_f32_16x16x4_f32(false, a, false, b3, (short)0, acc3, false, false);
    }

    // C/D layout: VGPR r -> M = r (+8 for lanes 16-31), N = lane%16
    const int mOff = (lane >> 4) * 8;
#pragma unroll
    for (int r = 0; r < 8; ++r) {
        float* o = H + (size_t)(rowBase + r + mOff) * HIDC + m;
        o[0]  = acc0[r];
        o[16] = acc1[r];
        o[32] = acc2[r];
        o[48] = acc3[r];
    }
}

// ---------------------------------------------------------------------------
// Edge scatter: agg[dst] += h[src] * edge_norm.  One wave32 per edge,
// 2 channels per lane (float2 gather, two f32 atomics -> contiguous 256B/wave).
// Working set (~100 MB) fits the 192 MB L2, so these RMWs are L2-resident.
// ---------------------------------------------------------------------------
__global__ void scatter_kernel(const float* __restrict__ H, const int* __restrict__ src,
                               const int* __restrict__ dst, const float* __restrict__ en,
                               float* __restrict__ agg, int E) {
    int t = blockIdx.x * blockDim.x + threadIdx.x;
    int e = t >> 5;
    if (e >= E) return;
    int lane = t & 31;
    int s = src[e];
    int d = dst[e];
    float w = en[e];
    int c = lane * 2;
    const float2 hv = *(const float2*)(H + (size_t)s * HIDC + c);
    float* o = agg + (size_t)d * HIDC + c;
    atomicAdd(o,     hv.x * w);
    atomicAdd(o + 1, hv.y * w);
}

// ---------------------------------------------------------------------------
// Fused self-loop + bias + ReLU:  agg = relu(agg + H*self_norm + b)
// ---------------------------------------------------------------------------
__global__ void finalize_kernel(float* __restrict__ agg, const float* __restrict__ H,
                                const float* __restrict__ self_norm,
                                const float* __restrict__ b, int N) {
    int t = blockIdx.x * blockDim.x + threadIdx.x;
    if (t >= N * HIDC) return;
    int i = t >> 6;
    int c = t & 63;
    float v = agg[t] + H[t] * self_norm[i] + b[c];
    agg[t] = fmaxf(v, 0.0f);
}

// ---------------------------------------------------------------------------
// Global mean pool (atomic accumulation of sums and counts)
// ---------------------------------------------------------------------------
__global__ void pool_kernel(const float* __restrict__ h, const int* __restrict__ batch,
                            float* __restrict__ sums, float* __restrict__ cnt, int N) {
    int t = blockIdx.x * blockDim.x + threadIdx.x;
    if (t >= N * HIDC) return;
    int i = t >> 6;
    int c = t & 63;
    int g = batch[i];
    atomicAdd(&sums[g * HIDC + c], h[t]);
    if (c == 0) atomicAdd(&cnt[g], 1.0f);
}

// ---------------------------------------------------------------------------
// Classifier: out[g, j] = mean_emb[g] @ Wc + bc   (64 x 10, one block)
// ---------------------------------------------------------------------------
__global__ void classify_kernel(const float* __restrict__ sums, const float* __restrict__ cnt,
                                const float* __restrict__ Wc, const float* __restrict__ bc,
                                float* __restrict__ out) {
    int t = threadIdx.x;
    if (t >= NGRAPH * NCLS) return;
    int g = t / NCLS;
    int j = t % NCLS;
    float inv = 1.0f / fmaxf(cnt[g], 1.0f);
    float acc = bc[j];
#pragma unroll
    for (int k = 0; k < HIDC; ++k)
        acc += sums[g * HIDC + k] * inv * Wc[k * NCLS + j];
    out[g * NCLS + j] = acc;
}

// ---------------------------------------------------------------------------
// Host launcher
// ---------------------------------------------------------------------------
extern "C" void kernel_launch(void* const* d_in, const int* in_sizes, int n_in,
                              void* d_out, int out_size, void* d_ws, size_t ws_size,
                              hipStream_t stream) {
    const float* x     = (const float*)d_in[0];
    const int*   eidx  = (const int*)d_in[1];
    const int*   batch = (const int*)d_in[2];
    const float* W1 = (const float*)d_in[3];
    const float* b1 = (const float*)d_in[4];
    const float* W2 = (const float*)d_in[5];
    const float* b2 = (const float*)d_in[6];
    const float* W3 = (const float*)d_in[7];
    const float* b3 = (const float*)d_in[8];
    const float* Wc = (const float*)d_in[9];
    const float* bc = (const float*)d_in[10];
    float* out = (float*)d_out;

    const int N = in_sizes[0] / HIDC;   // 100000
    const int E = in_sizes[1] / 2;      // 1600000
    const int* src = eidx;
    const int* dst = eidx + E;

    // workspace layout (floats)
    float* ws        = (float*)d_ws;
    float* bufH      = ws;                              // N*64
    float* bufA      = bufH + (size_t)N * HIDC;         // N*64
    float* dinv      = bufA + (size_t)N * HIDC;         // N  (deg, then dinv in place)
    float* self_norm = dinv + N;                        // N
    float* edge_norm = self_norm + N;                   // E
    float* psums     = edge_norm + E;                   // 64*64
    float* pcnt      = psums + NGRAPH * HIDC;           // 64

    const int TB = 256;
    auto blk = [](long n, int tb) { return (int)((n + tb - 1) / tb); };

    // --- normalization precompute ---
    fill0_kernel<<<blk(N, TB), TB, 0, stream>>>(dinv, N);
    deg_kernel<<<blk(E, TB), TB, 0, stream>>>(dst, dinv, E);
    dinv_kernel<<<blk(N, TB), TB, 0, stream>>>(dinv, self_norm, N);
    edge_norm_kernel<<<blk(E, TB), TB, 0, stream>>>(src, dst, dinv, edge_norm, E);

    const int gemmBlocks    = blk((long)(N / 16), 8);     // 8 row-tiles per block
    const long scatterThr   = (long)E * 32;
    const long nodeChanThr  = (long)N * HIDC;

    const float* Ws[3] = {W1, W2, W3};
    const float* bs[3] = {b1, b2, b3};

    for (int l = 0; l < 3; ++l) {
        const float* Xin = (l == 0) ? x : bufA;
        gemm64_wmma_kernel<<<gemmBlocks, TB, 0, stream>>>(Xin, Ws[l], bufH, N);
        fill0_kernel<<<blk(nodeChanThr, TB), TB, 0, stream>>>(bufA, (int)nodeChanThr);
        scatter_kernel<<<blk(scatterThr, TB), TB, 0, stream>>>(bufH, src, dst, edge_norm, bufA, E);
        finalize_kernel<<<blk(nodeChanThr, TB), TB, 0, stream>>>(bufA, bufH, self_norm, bs[l], N);
    }

    // --- pooling + classifier ---
    fill0_kernel<<<blk(NGRAPH * HIDC + NGRAPH, TB), TB, 0, stream>>>(psums, NGRAPH * HIDC + NGRAPH);
    pool_kernel<<<blk(nodeChanThr, TB), TB, 0, stream>>>(bufA, batch, psums, pcnt, N);
    classify_kernel<<<1, NGRAPH * NCLS, 0, stream>>>(psums, pcnt, Wc, bc, out);
}